// GNNEncoder_80367428042844
// MI455X (gfx1250) — compile-verified
//
#include <hip/hip_runtime.h>

typedef __attribute__((ext_vector_type(2))) float v2f;
typedef __attribute__((ext_vector_type(8))) float v8f;

// ---------------------------------------------------------------------------
// GEMM: H[N x 128] = X[N x 128] @ W[128 x 128], fp32 via V_WMMA_F32_16X16X4_F32
// One block = 256 threads = 8 waves; block covers 128 rows, wave covers 16 rows
// x full 128 output cols (8 accumulator tiles).
//
// W staged in LDS *transposed + XOR-swizzled*:  W[k][n] -> lwt[n*128 + (k ^ (4n & 63))]
// so each B fragment (W[k][n], W[k+1][n]) is one contiguous, bank-conflict-free
// ds_load_b64 directly into the even-aligned VGPR pair WMMA consumes.
// ---------------------------------------------------------------------------
__global__ void __launch_bounds__(256) gemm128_wmma_f32(
    const float* __restrict__ X, const float* __restrict__ W,
    float* __restrict__ H, int nRows)
{
    __shared__ float lwt[128 * 128];

    // Stage W (coalesced global float4 reads, transposed swizzled LDS writes).
    for (int i = threadIdx.x; i < 128 * 32; i += 256) {
        const int k  = i >> 5;          // W row (input dim), 32 float4 per row
        const int n4 = (i & 31) * 4;    // W col (output dim)
        const float4 w = *(const float4*)&W[k * 128 + n4];
        lwt[(n4 + 0) * 128 + (k ^ ((4 * (n4 + 0)) & 63))] = w.x;
        lwt[(n4 + 1) * 128 + (k ^ ((4 * (n4 + 1)) & 63))] = w.y;
        lwt[(n4 + 2) * 128 + (k ^ ((4 * (n4 + 2)) & 63))] = w.z;
        lwt[(n4 + 3) * 128 + (k ^ ((4 * (n4 + 3)) & 63))] = w.w;
    }
    __syncthreads();

    const int wave    = threadIdx.x >> 5;
    const int lane    = threadIdx.x & 31;
    const int rowBase = blockIdx.x * 128 + wave * 16;
    if (rowBase >= nRows) return;   // wave-uniform; EXEC stays all-1s for WMMA

    const int hi    = lane >> 4;    // 0: lanes 0-15, 1: lanes 16-31
    const int lcol  = lane & 15;
    const int khalf = hi * 2;       // A/B: low half-wave holds K=0,1; high holds K=2,3
    const int sw    = (4 * lcol) & 63;                    // swizzle constant (mult of 4)
    const float* xrow = X + (size_t)(rowBase + lcol) * 128;
    const float* lbase = lwt + lcol * 128;                // + nt*16*128 via immediates

    v8f acc[8] = {};

    #pragma unroll 4
    for (int kk = 0; kk < 128; kk += 4) {
        v2f a = *(const v2f*)(xrow + kk + khalf);         // A: 16x4 fragment (b64)
        const int kx = (kk + khalf) ^ sw;                 // even -> pair stays contiguous
        #pragma unroll
        for (int nt = 0; nt < 8; ++nt) {
            v2f b = *(const v2f*)(lbase + nt * (16 * 128) + kx);  // ds_load_b64
            acc[nt] = __builtin_amdgcn_wmma_f32_16x16x4_f32(
                false, a, false, b, (short)0, acc[nt], false, false);
        }
    }

    // C/D layout: VGPR r -> M = r + 8*hi, N = lcol
    const int rowOut = rowBase + 8 * hi;
    #pragma unroll
    for (int nt = 0; nt < 8; ++nt) {
        #pragma unroll
        for (int r = 0; r < 8; ++r) {
            H[(size_t)(rowOut + r) * 128 + nt * 16 + lcol] = acc[nt][r];
        }
    }
}

// ---------------------------------------------------------------------------
// Degree / normalization
// ---------------------------------------------------------------------------
__global__ void deg_init(float* __restrict__ deg, int n) {
    int i = blockIdx.x * 256 + threadIdx.x;
    if (i < n) deg[i] = 1.0f;                       // self-loop contribution
}

__global__ void deg_count(const int* __restrict__ colIdx, float* __restrict__ deg, int nE) {
    int e = blockIdx.x * 256 + threadIdx.x;
    if (e < nE) unsafeAtomicAdd(&deg[colIdx[e]], 1.0f);
}

__global__ void deg_rsqrt(const float* __restrict__ deg, float* __restrict__ dinv, int n) {
    int i = blockIdx.x * 256 + threadIdx.x;
    if (i < n) dinv[i] = rsqrtf(deg[i]);
}

// ---------------------------------------------------------------------------
// out[n][d] = bias[d]   (aggregation target init)
// ---------------------------------------------------------------------------
__global__ void __launch_bounds__(256) init_bias(
    float* __restrict__ out, const float* __restrict__ b, int nNodes)
{
    size_t idx = (size_t)blockIdx.x * 256 + threadIdx.x;   // one float4 / thread
    if (idx >= (size_t)nNodes * 32) return;
    ((float4*)out)[idx] = ((const float4*)b)[idx & 31];
}

// ---------------------------------------------------------------------------
// Edge scatter: out[col] += H[row] * dinv[row]*dinv[col]
// One wave per edge, one float4 of the 128-dim feature per lane:
// coalesced B128 gather + 4x GLOBAL_ATOMIC_ADD_F32 scatter (L2-resident).
// ---------------------------------------------------------------------------
__global__ void __launch_bounds__(256) scatter_edges(
    const float* __restrict__ H, const int* __restrict__ rowIdx,
    const int* __restrict__ colIdx, const float* __restrict__ dinv,
    float* __restrict__ out, int nEdges)
{
    const int wave = threadIdx.x >> 5;
    const int lane = threadIdx.x & 31;
    const int e = blockIdx.x * 8 + wave;
    if (e >= nEdges) return;
    const int r = rowIdx[e];
    const int c = colIdx[e];
    const float nrm = dinv[r] * dinv[c];
    const float4 hv = *(const float4*)(H + (size_t)r * 128 + lane * 4);
    float* op = out + (size_t)c * 128 + lane * 4;
    unsafeAtomicAdd(op + 0, hv.x * nrm);
    unsafeAtomicAdd(op + 1, hv.y * nrm);
    unsafeAtomicAdd(op + 2, hv.z * nrm);
    unsafeAtomicAdd(op + 3, hv.w * nrm);
}

// ---------------------------------------------------------------------------
// out[i] = relu(out[i] + H[i] * dinv[i]^2)   (self-loop term + ReLU)
// ---------------------------------------------------------------------------
__global__ void __launch_bounds__(256) selfloop_relu(
    float* __restrict__ out, const float* __restrict__ H,
    const float* __restrict__ dinv, int nNodes)
{
    size_t idx = (size_t)blockIdx.x * 256 + threadIdx.x;   // one float4 / thread
    if (idx >= (size_t)nNodes * 32) return;
    const int node = (int)(idx >> 5);
    const float d = dinv[node];
    const float d2 = d * d;
    float4 o = ((const float4*)out)[idx];
    const float4 hv = ((const float4*)H)[idx];
    o.x = fmaxf(fmaf(hv.x, d2, o.x), 0.0f);
    o.y = fmaxf(fmaf(hv.y, d2, o.y), 0.0f);
    o.z = fmaxf(fmaf(hv.z, d2, o.z), 0.0f);
    o.w = fmaxf(fmaf(hv.w, d2, o.w), 0.0f);
    ((float4*)out)[idx] = o;
}

// ---------------------------------------------------------------------------
extern "C" void kernel_launch(void* const* d_in, const int* in_sizes, int n_in,
                              void* d_out, int out_size, void* d_ws, size_t ws_size,
                              hipStream_t stream)
{
    const float* x  = (const float*)d_in[0];
    const int*   ei = (const int*)d_in[1];     // edge_index [2, E] (int32)
    const float* W0 = (const float*)d_in[2];
    const float* b0 = (const float*)d_in[3];
    const float* W1 = (const float*)d_in[4];
    const float* b1 = (const float*)d_in[5];
    const float* W2 = (const float*)d_in[6];
    const float* b2 = (const float*)d_in[7];
    float* out = (float*)d_out;

    const int N = in_sizes[0] / 128;
    const int E = in_sizes[1] / 2;
    const int* rowIdx = ei;        // edge_index[0] = source
    const int* colIdx = ei + E;    // edge_index[1] = target

    // Workspace: H (N*128 f32), dinv (N), deg (N)
    float* H    = (float*)d_ws;
    float* dinv = H + (size_t)N * 128;
    float* deg  = dinv + N;

    const dim3 blk(256);
    const int nodeBlocks = (N + 255) / 256;
    const int edgeTBlks  = (E + 255) / 256;
    const int gemmBlocks = (N + 127) / 128;
    const int vecBlocks  = (int)(((size_t)N * 32 + 255) / 256);
    const int edgeBlocks = (E + 7) / 8;

    deg_init <<<nodeBlocks, blk, 0, stream>>>(deg, N);
    deg_count<<<edgeTBlks,  blk, 0, stream>>>(colIdx, deg, E);
    deg_rsqrt<<<nodeBlocks, blk, 0, stream>>>(deg, dinv, N);

    const float* Ws[3] = {W0, W1, W2};
    const float* bs[3] = {b0, b1, b2};
    const float* in = x;
    for (int l = 0; l < 3; ++l) {
        gemm128_wmma_f32<<<gemmBlocks, blk, 0, stream>>>(in, Ws[l], H, N);
        init_bias       <<<vecBlocks,  blk, 0, stream>>>(out, bs[l], N);
        scatter_edges   <<<edgeBlocks, blk, 0, stream>>>(H, rowIdx, colIdx, dinv, out, E);
        selfloop_relu   <<<vecBlocks,  blk, 0, stream>>>(out, H, dinv, N);
        in = out;   // next layer reads aggregated output (stream-ordered)
    }
}